// AttnDecoderRNN_49813030699679
// MI455X (gfx1250) — compile-verified
//
#include <hip/hip_runtime.h>
#include <math.h>

// AttnDecoderRNN single step for MI455X (gfx1250, wave32).
// Bandwidth-bound GEMV chain; fp32 matrix pipe via V_WMMA_F32_16X16X4_F32,
// weights streamed with maximal-width global_load_b128.

#define VOCAB 50257
#define H     1024
#define L     128

typedef __attribute__((ext_vector_type(2))) float v2f;
typedef __attribute__((ext_vector_type(4))) float v4f;
typedef __attribute__((ext_vector_type(8))) float v8f;

// Workspace layout (floats):
//   [0     .. 2048)  xqh  = [q, h]
//   [2048  .. 4096)  xqa  = [q, attn_applied]
//   [4096  .. 5120)  gru_in (relu(combined))
//   [5120  .. 8192)  gi
//   [8192  ..11264)  gh
//   [11264 ..12288)  h_new
//   [12288]          log-softmax shift (max + log(sum))
#define WS_XQH   0
#define WS_XQA   2048
#define WS_GIN   4096
#define WS_GI    5120
#define WS_GH    8192
#define WS_HNEW  11264
#define WS_SHIFT 12288

// One wave computes out[row_base .. row_base+16) = W[rows] . x + bias.
// A fragment = x broadcast (all 16 M-rows identical), B fragment = 16 weight
// rows as columns. Fragment layout per CDNA5 ISA: lanes 0-15 supply K-slots
// 0,1 in v0,v1; lanes 16-31 supply slots 2,3. K-chunks commute under
// accumulation, so each lane loads a contiguous float4 (b128) that feeds two
// WMMAs: lanes 0-15 cover global k..k+3, lanes 16-31 cover k+4..k+7;
// WMMA#0 takes the .xy halves, WMMA#1 the .zw halves. Result D[0][n] sits in
// c[0] of lane n.
__device__ __forceinline__ void wmma_gemv16(
    const float* __restrict__ W, const float* __restrict__ x,
    const float* __restrict__ bias, float* __restrict__ out,
    int row_base, int n_rows, int K, bool relu)
{
    const int lane = threadIdx.x & 31;
    const int n    = lane & 15;
    const int kh   = lane >> 4;                 // 0: k..k+3, 1: k+4..k+7
    int row  = row_base + n;
    int rowc = row < n_rows ? row : (n_rows - 1);   // clamp for addressing only

    const v4f* __restrict__ wrow = (const v4f*)(W + (size_t)rowc * (size_t)K);
    const v4f* __restrict__ xv   = (const v4f*)x;

    v8f c = {0.f, 0.f, 0.f, 0.f, 0.f, 0.f, 0.f, 0.f};

    const int nk4 = K >> 2;                     // number of float4 chunks
    #pragma unroll 4
    for (int k4 = 0; k4 < nk4; k4 += 2) {       // each iter covers K += 8
        v4f a4 = xv[k4 + kh];                   // broadcast activation chunk
        v4f b4 = wrow[k4 + kh];                 // this lane's weight-row chunk
        v2f a0 = a4.xy, a1 = a4.zw;
        v2f b0 = b4.xy, b1 = b4.zw;
        c = __builtin_amdgcn_wmma_f32_16x16x4_f32(
                false, a0, false, b0, (short)0, c, false, false);
        c = __builtin_amdgcn_wmma_f32_16x16x4_f32(
                false, a1, false, b1, (short)0, c, false, false);
    }

    if (lane < 16 && row < n_rows) {
        float r = c[0] + bias[row];
        if (relu) r = r > 0.f ? r : 0.f;
        out[row] = r;
    }
}

// ---- stage 0: gather embedding row, stage [q,h] and [q,*] vectors ----------
__global__ void k_prep(const int* __restrict__ ids,
                       const float* __restrict__ hidden,
                       const float* __restrict__ emb,
                       float* __restrict__ ws)
{
    const int id = ids[0];
    const float* __restrict__ q = emb + (size_t)id * H;
    for (int j = threadIdx.x; j < H; j += blockDim.x) {
        float qv = q[j];
        ws[WS_XQH + j]       = qv;          // xqh[0:H] = q
        ws[WS_XQA + j]       = qv;          // xqa[0:H] = q
        ws[WS_XQH + H + j]   = hidden[j];   // xqh[H:2H] = h
    }
}

// ---- stage 1: attention scores (WMMA GEMV), softmax, weighted sum ----------
__global__ void k_attn(const float* __restrict__ attn_W,
                       const float* __restrict__ attn_b,
                       const float* __restrict__ enc,
                       float* __restrict__ ws,
                       float* __restrict__ d_out)
{
    __shared__ float sc[L];
    __shared__ float s_inv;

    const int wave = threadIdx.x >> 5;      // 8 waves, 16 rows each -> L=128
    wmma_gemv16(attn_W, ws + WS_XQH, attn_b, sc, wave * 16, L, 2 * H, false);
    __syncthreads();

    if (threadIdx.x == 0) {
        float mx = -3.402823466e38f;
        for (int l = 0; l < L; ++l) mx = fmaxf(mx, sc[l]);
        float s = 0.f;
        for (int l = 0; l < L; ++l) { float e = expf(sc[l] - mx); sc[l] = e; s += e; }
        s_inv = 1.f / s;
    }
    __syncthreads();

    for (int l = threadIdx.x; l < L; l += blockDim.x) {
        float w = sc[l] * s_inv;
        sc[l] = w;
        d_out[VOCAB + H + l] = w;           // attn_weights output
    }
    __syncthreads();

    // attn_applied[j] = sum_l w[l] * enc[l, j]  (coalesced over j)
    for (int j = threadIdx.x; j < H; j += blockDim.x) {
        float acc = 0.f;
        #pragma unroll 4
        for (int l = 0; l < L; ++l) acc += sc[l] * enc[l * H + j];
        ws[WS_XQA + H + j] = acc;
    }
}

// ---- stage 2: combined = relu([q,attn] @ comb_W.T + b) ---------------------
__global__ void k_comb(const float* __restrict__ comb_W,
                       const float* __restrict__ comb_b,
                       float* __restrict__ ws)
{
    const int tile = blockIdx.x * 8 + (threadIdx.x >> 5);
    wmma_gemv16(comb_W, ws + WS_XQA, comb_b, ws + WS_GIN, tile * 16, H, 2 * H, true);
}

// ---- stage 3: GRU gate GEMVs (gi, gh) --------------------------------------
__global__ void k_gates(const float* __restrict__ W_ih,
                        const float* __restrict__ W_hh,
                        const float* __restrict__ b_ih,
                        const float* __restrict__ b_hh,
                        float* __restrict__ ws)
{
    const int tile = blockIdx.x * 8 + (threadIdx.x >> 5);
    if (blockIdx.y == 0)
        wmma_gemv16(W_ih, ws + WS_GIN,     b_ih, ws + WS_GI, tile * 16, 3 * H, H, false);
    else
        wmma_gemv16(W_hh, ws + WS_XQH + H, b_hh, ws + WS_GH, tile * 16, 3 * H, H, false);
}

// ---- stage 4: GRU elementwise gates ----------------------------------------
__global__ void k_gru(float* __restrict__ ws, float* __restrict__ d_out)
{
    const float* __restrict__ gi = ws + WS_GI;
    const float* __restrict__ gh = ws + WS_GH;
    const float* __restrict__ h  = ws + WS_XQH + H;
    for (int j = blockIdx.x * blockDim.x + threadIdx.x; j < H;
         j += blockDim.x * gridDim.x) {
        float r  = 1.f / (1.f + expf(-(gi[j]         + gh[j])));
        float z  = 1.f / (1.f + expf(-(gi[H + j]     + gh[H + j])));
        float nn = tanhf(gi[2 * H + j] + r * gh[2 * H + j]);
        float hn = (1.f - z) * nn + z * h[j];
        ws[WS_HNEW + j]  = hn;
        d_out[VOCAB + j] = hn;              // hidden output
    }
}

// ---- stage 5: output projection (dominant, ~206 MB streamed once) ----------
__global__ void k_logits(const float* __restrict__ out_W,
                         const float* __restrict__ out_b,
                         const float* __restrict__ ws,
                         float* __restrict__ d_out)
{
    const int tile = blockIdx.x * 8 + (threadIdx.x >> 5);
    wmma_gemv16(out_W, ws + WS_HNEW, out_b, d_out, tile * 16, VOCAB, H, false);
}

// ---- stage 6: log-softmax shift (single-block reduction) -------------------
__global__ void k_lse(const float* __restrict__ logits, float* __restrict__ ws)
{
    __shared__ float red[1024];
    float mx = -3.402823466e38f;
    for (int i = threadIdx.x; i < VOCAB; i += blockDim.x)
        mx = fmaxf(mx, logits[i]);
    red[threadIdx.x] = mx;
    __syncthreads();
    for (int s = blockDim.x >> 1; s > 0; s >>= 1) {
        if (threadIdx.x < s) red[threadIdx.x] = fmaxf(red[threadIdx.x], red[threadIdx.x + s]);
        __syncthreads();
    }
    mx = red[0];
    __syncthreads();
    float sum = 0.f;
    for (int i = threadIdx.x; i < VOCAB; i += blockDim.x)
        sum += expf(logits[i] - mx);
    red[threadIdx.x] = sum;
    __syncthreads();
    for (int s = blockDim.x >> 1; s > 0; s >>= 1) {
        if (threadIdx.x < s) red[threadIdx.x] += red[threadIdx.x + s];
        __syncthreads();
    }
    if (threadIdx.x == 0) ws[WS_SHIFT] = mx + logf(red[0]);
}

// ---- stage 7: apply shift ---------------------------------------------------
__global__ void k_finish(float* __restrict__ d_out, const float* __restrict__ ws)
{
    const float shift = ws[WS_SHIFT];
    const int i = blockIdx.x * blockDim.x + threadIdx.x;
    if (i < VOCAB) d_out[i] -= shift;
}

extern "C" void kernel_launch(void* const* d_in, const int* in_sizes, int n_in,
                              void* d_out, int out_size, void* d_ws, size_t ws_size,
                              hipStream_t stream)
{
    const int*   ids    = (const int*)  d_in[0];
    const float* hidden = (const float*)d_in[1];
    const float* enc    = (const float*)d_in[2];
    const float* emb    = (const float*)d_in[3];
    const float* attn_W = (const float*)d_in[4];
    const float* attn_b = (const float*)d_in[5];
    const float* comb_W = (const float*)d_in[6];
    const float* comb_b = (const float*)d_in[7];
    const float* W_ih   = (const float*)d_in[8];
    const float* W_hh   = (const float*)d_in[9];
    const float* b_ih   = (const float*)d_in[10];
    const float* b_hh   = (const float*)d_in[11];
    const float* out_W  = (const float*)d_in[12];
    const float* out_b  = (const float*)d_in[13];
    float* out = (float*)d_out;
    float* ws  = (float*)d_ws;

    (void)in_sizes; (void)n_in; (void)out_size; (void)ws_size;

    k_prep  <<<1,           256, 0, stream>>>(ids, hidden, emb, ws);
    k_attn  <<<1,           256, 0, stream>>>(attn_W, attn_b, enc, ws, out);
    k_comb  <<<8,           256, 0, stream>>>(comb_W, comb_b, ws);     // 64 tiles
    k_gates <<<dim3(24, 2), 256, 0, stream>>>(W_ih, W_hh, b_ih, b_hh, ws); // 192 tiles x2
    k_gru   <<<4,           256, 0, stream>>>(ws, out);
    k_logits<<<393,         256, 0, stream>>>(out_W, out_b, ws, out);  // 3144 tiles
    k_lse   <<<1,          1024, 0, stream>>>(out, ws);
    k_finish<<<197,         256, 0, stream>>>(out, ws);
}